// SelfAdderConvReLUBlock_16209206575286
// MI455X (gfx1250) — compile-verified
//
#include <hip/hip_runtime.h>
#include <hip/hip_bf16.h>
#include <stdint.h>

// AdderNet conv+ReLU for MI455X (gfx1250).
// L1 distance is not bilinear -> WMMA inapplicable; workload is VALU-bound.
// CDNA5 features used: TDM tensor_load_to_lds (iterate+pad modes) staging x
// into 320KB LDS, s_wait_tensorcnt, wave32-shaped 256-thread blocks, scalar
// (uniform) weight loads co-issued with a 2-VALU-per-term inner loop.

typedef unsigned int v4u __attribute__((ext_vector_type(4)));
typedef int          v8i __attribute__((ext_vector_type(8)));
typedef int          v4i __attribute__((ext_vector_type(4)));

#define N_ 8
#define C_ 64
#define H_ 128
#define W_ 128
#define O_ 64
#define LDSW 132                    // 128 cols + 1 left pad + 3 pad/align
#define LDSROWS 6                   // 4 output rows + 2 halo rows
#define CH_STRIDE (LDSROWS * LDSW)  // 792 floats per channel
#define SMEM_FLOATS (C_ * CH_STRIDE)
#define SMEM_BYTES (SMEM_FLOATS * 4)  // 202752 B < 320KB WGP LDS

__global__ __launch_bounds__(256, 1) void adder_conv_relu_kernel(
    const float* __restrict__ x, const float* __restrict__ w,
    const float* __restrict__ b, float* __restrict__ out) {
  extern __shared__ float smem[];
  const int tid = threadIdx.x;
  const int col = tid & 127;   // output column
  const int ty  = tid >> 7;    // 0..1; thread handles rows rbase+ty, rbase+ty+2
  const int rb  = blockIdx.x;  // 0..31 row blocks of 4 rows
  const int ob  = blockIdx.y;  // 0..3 blocks of 16 output channels
  const int n   = blockIdx.z;  // batch
  const int rbase = rb * 4;

  const int top     = (rbase == 0);
  const int bot     = (rbase == H_ - 4);
  const int nrows   = LDSROWS - top - bot;  // valid input rows to DMA
  const int ldsRow0 = top ? 1 : 0;          // LDS row where DMA starts
  const int gRow0   = top ? 0 : (rbase - 1);

  // Zero horizontal pad columns L=0 (input col -1) and L=129 (input col 128).
  // Disjoint from TDM-written addresses (TDM writes L=1..128; pad dwords skipped).
  for (int i = tid; i < C_ * LDSROWS * 2; i += 256) {
    int c   = i / (LDSROWS * 2);
    int rem = i - c * (LDSROWS * 2);
    int r   = rem >> 1;
    int L   = (rem & 1) ? 129 : 0;
    smem[c * CH_STRIDE + r * LDSW + L] = 0.0f;
  }
  // Zero the vertical padding row at image top/bottom edges.
  if (top) {
    for (int i = tid; i < C_ * LDSW; i += 256) {
      int c = i / LDSW;
      smem[c * CH_STRIDE + (i - c * LDSW)] = 0.0f;  // LDS row 0 == input row -1
    }
  }
  if (bot) {
    for (int i = tid; i < C_ * LDSW; i += 256) {
      int c = i / LDSW;
      smem[c * CH_STRIDE + 5 * LDSW + (i - c * LDSW)] = 0.0f;  // input row 128
    }
  }

  // --- Tensor Data Mover: stage x[n, 0..63, gRow0..gRow0+nrows-1, 0..127] ---
  // 2D tile (128 cols x nrows rows), pad 4 dwords every 128 dwords (row pitch
  // 132 floats), iterate 64x over channels with fixed LDS stride 792 floats.
  if (tid == 0) {
    uint64_t gaddr = (uint64_t)(uintptr_t)(x + ((size_t)n * C_ * H_ + gRow0) * W_);
    uint32_t lds_base  = (uint32_t)(uintptr_t)smem;
    uint32_t lds_start = lds_base + (uint32_t)((ldsRow0 * LDSW + 1) * 4);

    v4u g0;
    g0[0] = 1u;                 // count=1 valid user descriptor
    g0[1] = lds_start;          // lds_addr (bytes)
    g0[2] = (uint32_t)gaddr;    // global_addr[31:0]
    g0[3] = (uint32_t)((gaddr >> 32) & 0x1ffffffu) | (2u << 30);  // [56:32], type=2

    v8i g1;
    // data_size=2 (4B), pad_enable=1, pad_interval=6 (128 dwords), pad_amount=3 (4 dwords)
    g1[0] = (int)((2u << 16) | (1u << 20) | (6u << 22) | (3u << 25));
    g1[1] = (int)((uint32_t)W_ << 16);        // tensor_dim0[15:0] @ bits[63:48]
    g1[2] = (int)((uint32_t)nrows << 16);     // tensor_dim1[15:0] @ bits[95:80]
    g1[3] = (int)((uint32_t)W_ << 16);        // tile_dim0 @ bits[127:112]
    g1[4] = (int)((uint32_t)nrows);           // tile_dim1 @ [143:128], tile_dim2=0
    g1[5] = (int)W_;                          // tensor_dim0_stride low32 = 128
    g1[6] = (int)((uint32_t)(H_ * W_) << 16); // tensor_dim1_stride[15:0] @ [223:208]
    g1[7] = (int)((uint32_t)(H_ * W_) >> 16); // tensor_dim1_stride[47:16]

    v4i g2;
    g2[0] = C_;                 // tensor_dim2 (unused in iterate mode)
    g2[1] = CH_STRIDE;          // lds_addr_increment per iteration (data_size units)
    g2[2] = H_ * W_;            // global_addr_increment per iteration = 16384 elems
    g2[3] = (C_ - 1) << 16;     // iterate_count=63 -> 64 iterations @ bits[127:112]

    v4i g3 = {0, 0, 0, 0};
    v8i gx = {0, 0, 0, 0, 0, 0, 0, 0};  // extra arg on clang-23 toolchain; zero-filled
    __builtin_amdgcn_tensor_load_to_lds(g0, g1, g2, g3, gx, 0);
  }
  __builtin_amdgcn_s_wait_tensorcnt(0);
  __syncthreads();

  // --- Compute: 16 output channels x 2 pixels per thread ---
  float acc0[16], acc1[16];
#pragma unroll
  for (int i = 0; i < 16; ++i) { acc0[i] = 0.0f; acc1[i] = 0.0f; }

  for (int c = 0; c < C_; ++c) {
    const float* xc = smem + c * CH_STRIDE;
    // 15 LDS loads: rows ty..ty+4, cols (col..col+2) in padded L coords.
    float xv[5][3];
#pragma unroll
    for (int r = 0; r < 5; ++r) {
#pragma unroll
      for (int dc = 0; dc < 3; ++dc) {
        xv[r][dc] = xc[(ty + r) * LDSW + col + dc];
      }
    }
#pragma unroll
    for (int o16 = 0; o16 < 16; ++o16) {
      // Uniform index -> scalar loads on the SMEM pipe, broadcast to all lanes.
      const float* wr = w + ((size_t)(ob * 16 + o16) * C_ + c) * 9;
#pragma unroll
      for (int kh = 0; kh < 3; ++kh) {
#pragma unroll
        for (int kw = 0; kw < 3; ++kw) {
          const float wv = wr[kh * 3 + kw];
          acc0[o16] += fabsf(xv[kh][kw]     - wv);  // v_sub + v_add |src|
          acc1[o16] += fabsf(xv[kh + 2][kw] - wv);
        }
      }
    }
  }

  // --- Bias, ReLU, store ---
  const int orow0 = rbase + ty;  // second pixel is orow0 + 2
#pragma unroll
  for (int o16 = 0; o16 < 16; ++o16) {
    const int o = ob * 16 + o16;
    const float bv = b[o];  // uniform -> scalar load
    const float v0 = fmaxf(bv - acc0[o16], 0.0f);
    const float v1 = fmaxf(bv - acc1[o16], 0.0f);
    const size_t base = ((size_t)(n * O_ + o) * H_ + orow0) * W_ + col;
    out[base]          = v0;
    out[base + 2 * W_] = v1;
  }
}

extern "C" void kernel_launch(void* const* d_in, const int* in_sizes, int n_in,
                              void* d_out, int out_size, void* d_ws, size_t ws_size,
                              hipStream_t stream) {
  (void)in_sizes; (void)n_in; (void)d_ws; (void)ws_size; (void)out_size;
  const float* x = (const float*)d_in[0];
  const float* w = (const float*)d_in[1];
  const float* b = (const float*)d_in[2];
  float* out = (float*)d_out;

  // Opt in to >64KB dynamic LDS (idempotent; safe under graph capture).
  (void)hipFuncSetAttribute((const void*)adder_conv_relu_kernel,
                            hipFuncAttributeMaxDynamicSharedMemorySize, SMEM_BYTES);

  dim3 grid(H_ / 4, O_ / 16, N_);  // 32 x 4 x 8 = 1024 blocks
  adder_conv_relu_kernel<<<grid, 256, SMEM_BYTES, stream>>>(x, w, b, out);
}